// CircuitTorchWrapper_89008902242951
// MI455X (gfx1250) — compile-verified
//
#include <hip/hip_runtime.h>

// ---------------------------------------------------------------------------
// 20-qubit Ry/CNOT-chain circuit, 4 layers, on MI455X (gfx1250, wave32).
//
//  * Ry and CNOT are real -> real & imag float32 vectors evolve independently.
//  * The full CNOT chain == Gray permutation new[y] = old[y ^ (y>>1)], fused
//    into the store of the low-bits pass (stays inside aligned 2^11 blocks).
//  * Ry on qubits 16..19 and 12..15 fused into 16x16 matrices applied with
//    V_WMMA_F32_16X16X4_F32 (4 accumulating K=4 chunks per 16x16x16 product).
//  * Working set 8-16 MB -> fully L2 resident (192 MB); 8 kernel sweeps total.
//  * Trig via branch-free Cody-Waite sincos (no libm Payne-Hanek SALU bloat).
// ---------------------------------------------------------------------------

#define NQ     20
#define DIM    (1u << NQ)

typedef float v2f __attribute__((ext_vector_type(2)));
typedef float v8f __attribute__((ext_vector_type(8)));

// Branch-free sincos for |x| <= ~100 (here |x| < ~3): round-to-quadrant +
// two-term Cody-Waite reduction + minimax polynomials. ~1 ulp in fp32.
__device__ __forceinline__ void fast_sincos(float x, float* sOut, float* cOut) {
    const float fn = rintf(x * 0.636619772f);          // round(x / (pi/2))
    float r = fmaf(-fn, 1.57079637e+0f, x);            // x - n*pio2_hi
    r = fmaf(fn, 4.37113883e-8f, r);                   // - n*pio2_lo
    const float r2 = r * r;
    // sin(r), cos(r) on [-pi/4, pi/4]
    float sp = fmaf(r2, fmaf(r2, fmaf(r2, 2.75255622e-6f, -1.98408739e-4f),
                              8.33332930e-3f), -1.66666672e-1f);
    sp = fmaf(r * r2, sp, r);
    float cp = fmaf(r2, fmaf(r2, fmaf(r2, 2.44331571e-5f, -1.38867637e-3f),
                              4.16666418e-2f), -5.0e-1f);
    cp = fmaf(r2, cp, 1.0f);
    const int q = ((int)fn) & 3;
    float s = (q & 1) ? cp : sp;
    float c = (q & 1) ? sp : cp;
    if (q & 2)       s = -s;                           // quadrants 2,3
    if ((q + 1) & 2) c = -c;                           // quadrants 1,2
    *sOut = s;
    *cOut = c;
}

// Ry matrix element factor: Ry = [[c,-s],[s,c]]
__device__ __forceinline__ float ry_fac(float c, float s, int ib, int jb) {
    return (ib == jb) ? c : (ib ? s : -s);
}

// U = Ry(q0) (x) Ry(q1) (x) Ry(q2) (x) Ry(q3), element [i][j]; c4/s4 indexed
// only by unrolled constants (stays in registers, no scalarization ladders).
__device__ __forceinline__ float u_elem(const float* c4, const float* s4, int i, int j) {
    float u = 1.0f;
#pragma unroll
    for (int t = 0; t < 4; ++t) {
        int ib = (i >> (3 - t)) & 1;
        int jb = (j >> (3 - t)) & 1;
        u *= ry_fac(c4[t], s4[t], ib, jb);
    }
    return u;
}

// ---------------------------------------------------------------------------
// Pass H: Ry on qubits 0..9 (bits 19..10). Tile = 1024 hi-values x 8 lo-values
// (strided by 1024), held in 32 KB LDS. grid = (128 lo-tiles, 2 arrays).
// ---------------------------------------------------------------------------
__global__ __launch_bounds__(256)
void pass_hi(const float* __restrict__ theta,
             const float* __restrict__ srcRe,
             const float* __restrict__ srcIm,
             float* __restrict__ dstBase,
             int layer)
{
    __shared__ float S[1024 * 8];
    const int tid    = threadIdx.x;
    const int loBase = blockIdx.x * 8;
    const float* src = (blockIdx.y == 0) ? srcRe : srcIm;
    float* dst       = dstBase + (size_t)blockIdx.y * DIM;

    for (int i = tid; i < 8192; i += 256) {
        int hi = i >> 3, c = i & 7;
        S[i] = src[(hi << 10) + loBase + c];
    }
    __syncthreads();

    for (int q = 0; q < 10; ++q) {
        const int hb = 9 - q;                     // qubit q <-> hi-bit (9-q)
        float cc, ss;
        fast_sincos(theta[layer * NQ + q] * 0.5f, &ss, &cc);
        const int mask = (1 << hb) - 1;
        for (int p = tid; p < 4096; p += 256) {   // 512 pairs x 8 lo columns
            int c  = p & 7;
            int pr = p >> 3;
            int hi0 = ((pr >> hb) << (hb + 1)) | (pr & mask);
            int i0  = (hi0 << 3) + c;
            int i1  = i0 + (8 << hb);
            float a0 = S[i0], a1 = S[i1];
            S[i0] = cc * a0 - ss * a1;
            S[i1] = ss * a0 + cc * a1;
        }
        __syncthreads();
    }

    for (int i = tid; i < 8192; i += 256) {
        int hi = i >> 3, c = i & 7;
        dst[(hi << 10) + loBase + c] = S[i];
    }
}

// ---------------------------------------------------------------------------
// Pass L: Ry on qubits 10..19 (bits 9..0) + fused CNOT-chain Gray gather.
// Tile = 2^11 contiguous floats (8 KB LDS). grid = (512 tiles, 2 arrays).
//   qubits 10,11 (bits 9,8): scalar pair rotations
//   qubits 16..19 (bits 3..0): WMMA stage 0 (128 groups, 16 per wave)
//   qubits 12..15 (bits 7..4): WMMA stage 1 (8 matrices, 1 per wave)
// Both stages address LDS as  idx = base + e*stride  per lane:
//   stage0: base = ((wave*16+r)<<4), stride = 1
//   stage1: base = wave*256 + r,     stride = 16
// Store: dst[ybase+t] = S[(t ^ (t>>1)) ^ (ybase_bit11 << 10)]
// ---------------------------------------------------------------------------
__global__ __launch_bounds__(256)
void pass_lo(const float* __restrict__ theta,
             const float* __restrict__ srcBase,
             float* __restrict__ dstBase,
             int layer)
{
    __shared__ float S[2048];
    const int tid = threadIdx.x;
    const unsigned ybase = (unsigned)blockIdx.x << 11;   // low 11 bits zero
    const float* src = srcBase + (size_t)blockIdx.y * DIM;
    float* dst       = dstBase + (size_t)blockIdx.y * DIM;

    // Input block for the Gray gather: bits >=11 of y^(y>>1)
    const unsigned xblock = (ybase ^ (ybase >> 1)) & ~2047u;
    for (int i = tid; i < 2048; i += 256) S[i] = src[xblock + i];
    __syncthreads();

    // --- scalar Ry on qubit 10 (bit 9) and qubit 11 (bit 8) ---
    for (int q = 10; q <= 11; ++q) {
        const int b = 19 - q;
        float cc, ss;
        fast_sincos(theta[layer * NQ + q] * 0.5f, &ss, &cc);
        const int mask = (1 << b) - 1;
        for (int p = tid; p < 1024; p += 256) {
            int i0 = ((p >> b) << (b + 1)) | (p & mask);
            int i1 = i0 | (1 << b);
            float a0 = S[i0], a1 = S[i1];
            S[i0] = cc * a0 - ss * a1;
            S[i1] = ss * a0 + cc * a1;
        }
        __syncthreads();
    }

    // --- two fused-4-qubit 16x16 WMMA stages ---
    const int lane = tid & 31;
    const int wave = tid >> 5;     // 0..7
    const int r    = lane & 15;    // A row / B,D column for this lane
    const int h    = lane >> 4;    // lane half

#pragma unroll
    for (int stage = 0; stage < 2; ++stage) {
        const int qBase  = stage ? 12 : 16;                 // qubit group
        const int base   = stage ? (wave * 256 + r)         // e*16 + n  layout
                                 : ((wave * 16 + r) << 4);  // n*16 + e  layout
        const int stride = stage ? 16 : 1;

        float c4[4], s4[4];
#pragma unroll
        for (int t = 0; t < 4; ++t)
            fast_sincos(theta[layer * NQ + qBase + t] * 0.5f, &s4[t], &c4[t]);

        v8f acc = {0.f, 0.f, 0.f, 0.f, 0.f, 0.f, 0.f, 0.f};
#pragma unroll
        for (int kc = 0; kc < 4; ++kc) {
            const int k0 = kc * 4 + 2 * h;
            // A 16x4 f32 chunk (ISA 7.12.2): lanes 0-15 -> K=k0,k0+1 of row r;
            // lanes 16-31 -> K=k0+2,k0+3 (h folds into k0).
            v2f a, b;
            a.x = u_elem(c4, s4, r, k0);
            a.y = u_elem(c4, s4, r, k0 + 1);
            // B 4x16 chunk: VGPR j holds row (j + 2h), column n = lane%16.
            // Direct LDS reads (ds_load) -- no private-array dynamic indexing.
            b.x = S[base + k0 * stride];
            b.y = S[base + (k0 + 1) * stride];
            acc = __builtin_amdgcn_wmma_f32_16x16x4_f32(
                      /*neg_a=*/false, a, /*neg_b=*/false, b,
                      /*c_mod=*/(short)0, acc,
                      /*reuse_a=*/false, /*reuse_b=*/false);
        }

        // D 16x16 f32 layout: VGPR v -> row v (lanes 0-15) / row v+8 (16-31).
        // Each wave writes only its own 256-float slice; barrier between stages.
#pragma unroll
        for (int v = 0; v < 8; ++v)
            S[base + (v + 8 * h) * stride] = acc[v];
        __syncthreads();
    }

    // --- fused CNOT chain: Gray-code gather on store ---
    const unsigned ybit11 = (ybase >> 11) & 1u;
    for (int t = tid; t < 2048; t += 256) {
        unsigned xl = ((unsigned)t ^ ((unsigned)t >> 1)) ^ (ybit11 << 10);
        dst[ybase + t] = S[xl];
    }
}

// ---------------------------------------------------------------------------
// Launch: 2 passes per layer, ping-pong OUT <-> WS so the final result lands
// in d_out (real[0..DIM) then imag[DIM..2*DIM), matching stack([re, im])).
// ---------------------------------------------------------------------------
extern "C" void kernel_launch(void* const* d_in, const int* in_sizes, int n_in,
                              void* d_out, int out_size, void* d_ws, size_t ws_size,
                              hipStream_t stream)
{
    (void)in_sizes; (void)n_in; (void)out_size; (void)ws_size;
    const float* theta = (const float*)d_in[0];   // (4, 20) float32
    const float* q0re  = (const float*)d_in[1];   // (DIM,)  float32
    const float* q0im  = (const float*)d_in[2];   // (DIM,)  float32
    float* OUT = (float*)d_out;                   // 2*DIM floats
    float* WS  = (float*)d_ws;                    // >= 2*DIM floats scratch

    const dim3 blk(256);
    const dim3 gh(128, 2);   // pass_hi: 128 lo-tiles x {re, im}
    const dim3 gl(512, 2);   // pass_lo: 512 tiles   x {re, im}

    // layer 0
    pass_hi<<<gh, blk, 0, stream>>>(theta, q0re, q0im, OUT, 0);
    pass_lo<<<gl, blk, 0, stream>>>(theta, OUT, WS, 0);
    // layer 1
    pass_hi<<<gh, blk, 0, stream>>>(theta, WS, WS + DIM, WS, 1);
    pass_lo<<<gl, blk, 0, stream>>>(theta, WS, OUT, 1);
    // layer 2
    pass_hi<<<gh, blk, 0, stream>>>(theta, OUT, OUT + DIM, OUT, 2);
    pass_lo<<<gl, blk, 0, stream>>>(theta, OUT, WS, 2);
    // layer 3
    pass_hi<<<gh, blk, 0, stream>>>(theta, WS, WS + DIM, WS, 3);
    pass_lo<<<gl, blk, 0, stream>>>(theta, WS, OUT, 3);
}